// Net_41901700940246
// MI455X (gfx1250) — compile-verified
//
#include <hip/hip_runtime.h>
#include <hip/hip_bf16.h>

typedef __attribute__((ext_vector_type(2))) float v2f;
typedef __attribute__((ext_vector_type(8))) float v8f;

#define F_DIM 128
#define H_DIM 256
#define C_DIM 40
#define BN_EPS 1e-5f

// ---------------------------------------------------------------------------
// Fused MLP head: h0 = ReLU(BN(x@W1 + b1)) @ W2 + b2   -> [N, 40] fp32
// One block = 16 output rows, 8 waves (256 threads, wave32).
// GEMM via V_WMMA_F32_16X16X4_F32 (fp32 WMMA, preserves reference precision).
// A-frag: lane L<16 -> row M=L, K = k + {0,1}; lane>=16 -> K = k + {2,3}.
// B-frag: N = lane%16, K = k + 2*(lane/16) + {0,1}.
// C/D  : vgpr r -> M = r + 8*(lane/16), N = lane%16.
// ---------------------------------------------------------------------------
__global__ __launch_bounds__(256) void mlp_kernel(
    const float* __restrict__ x, const float* __restrict__ W1,
    const float* __restrict__ b1, const float* __restrict__ gamma,
    const float* __restrict__ beta, const float* __restrict__ mean,
    const float* __restrict__ var, const float* __restrict__ W2,
    const float* __restrict__ b2, float* __restrict__ h0, int N)
{
    __shared__ float As[16 * 129];   // padded stride -> bank rotation
    __shared__ float Hs[16 * 257];

    const int tid  = threadIdx.x;
    const int wave = tid >> 5;
    const int lane = tid & 31;
    const int half = lane >> 4;
    const int l16  = lane & 15;
    const int row0 = blockIdx.x * 16;

    // cooperative, coalesced load of the 16x128 x-tile
    #pragma unroll
    for (int i = 0; i < 8; ++i) {
        int idx = tid + i * 256;
        int r = idx >> 7, c = idx & 127;
        int gr = row0 + r;
        As[r * 129 + c] = (gr < N) ? x[(long)gr * F_DIM + c] : 0.0f;
    }
    __syncthreads();

    // GEMM1: each wave owns 2 of the 16 column tiles of H=256
    #pragma unroll
    for (int t = 0; t < 2; ++t) {
        const int ct = wave * 2 + t;
        const int n  = ct * 16 + l16;
        v8f acc = {};
        for (int k = 0; k < F_DIM; k += 4) {
            v2f a, b;
            a.x = As[l16 * 129 + k + 2 * half + 0];
            a.y = As[l16 * 129 + k + 2 * half + 1];
            b.x = W1[(long)(k + 2 * half + 0) * H_DIM + n];
            b.y = W1[(long)(k + 2 * half + 1) * H_DIM + n];
            acc = __builtin_amdgcn_wmma_f32_16x16x4_f32(
                false, a, false, b, (short)0, acc, false, false);
        }
        // epilogue: bias + BatchNorm(eval) + ReLU -> LDS hidden tile
        const float bb = b1[n];
        const float g  = gamma[n], be = beta[n];
        const float mu = mean[n],  iv = rsqrtf(var[n] + BN_EPS);
        #pragma unroll
        for (int r = 0; r < 8; ++r) {
            const int m = r + 8 * half;
            float v = acc[r] + bb;
            v = (v - mu) * iv * g + be;
            Hs[m * 257 + n] = fmaxf(v, 0.0f);
        }
    }
    __syncthreads();

    // GEMM2: 16x40 output padded to 3 column tiles of 16; waves 0..2
    if (wave < 3) {
        const int n   = wave * 16 + l16;
        const bool nok = (n < C_DIM);
        v8f acc = {};
        for (int k = 0; k < H_DIM; k += 4) {
            v2f a, b;
            a.x = Hs[l16 * 257 + k + 2 * half + 0];
            a.y = Hs[l16 * 257 + k + 2 * half + 1];
            b.x = nok ? W2[(long)(k + 2 * half + 0) * C_DIM + n] : 0.0f;
            b.y = nok ? W2[(long)(k + 2 * half + 1) * C_DIM + n] : 0.0f;
            acc = __builtin_amdgcn_wmma_f32_16x16x4_f32(
                false, a, false, b, (short)0, acc, false, false);
        }
        const float bb = nok ? b2[n] : 0.0f;
        #pragma unroll
        for (int r = 0; r < 8; ++r) {
            const int m  = r + 8 * half;
            const int gr = row0 + m;
            if (nok && gr < N) h0[(long)gr * C_DIM + n] = acc[r] + bb;
        }
    }
}

// ---------------------------------------------------------------------------
__global__ void zero_kernel(float* __restrict__ p, long n)
{
    long i = (long)blockIdx.x * blockDim.x + threadIdx.x;
    if (i < n) p[i] = 0.0f;
}

// One edge per 8 lanes; lanes cover c = l + 8j (coalesced 40-float row).
// prev table (27MB) is L2-resident on MI455X (192MB L2) -> atomics stay on-chip.
__global__ void scatter_kernel(const int* __restrict__ src,
                               const int* __restrict__ dst,
                               const float* __restrict__ norm,
                               const float* __restrict__ prev,
                               float* __restrict__ next, int E)
{
    long tid = (long)blockIdx.x * blockDim.x + threadIdx.x;
    int e = (int)(tid >> 3);
    int l = (int)(tid & 7);
    if (e >= E) return;
    const int   s = src[e];
    const int   d = dst[e];
    const float w = norm[e];
    const float* pr = prev + (long)s * C_DIM;
    float*       nx = next + (long)d * C_DIM;
    #pragma unroll
    for (int j = 0; j < 5; ++j) {
        const int c = l + 8 * j;
        atomicAdd(&nx[c], w * pr[c]);
    }
}

// Fused hop-attention combine: acc += sigmoid(v . proj_w + proj_b) * v
__global__ void combine_kernel(const float* __restrict__ v,
                               float* __restrict__ acc,
                               const float* __restrict__ pw,
                               const float* __restrict__ pb, int N)
{
    int n = blockIdx.x * blockDim.x + threadIdx.x;
    if (n >= N) return;
    const float* row = v + (long)n * C_DIM;
    float d = pb[0];
    #pragma unroll
    for (int c = 0; c < C_DIM; ++c) d += row[c] * pw[c];
    const float r = 1.0f / (1.0f + expf(-d));
    float* ar = acc + (long)n * C_DIM;
    #pragma unroll
    for (int c = 0; c < C_DIM; ++c) ar[c] += r * row[c];
}

__global__ void lsm_kernel(const float* __restrict__ acc,
                           float* __restrict__ out, int N)
{
    int n = blockIdx.x * blockDim.x + threadIdx.x;
    if (n >= N) return;
    const float* row = acc + (long)n * C_DIM;
    float m = -3.402823466e38f;
    #pragma unroll
    for (int c = 0; c < C_DIM; ++c) m = fmaxf(m, row[c]);
    float s = 0.0f;
    #pragma unroll
    for (int c = 0; c < C_DIM; ++c) s += expf(row[c] - m);
    const float ls = logf(s);
    float* o = out + (long)n * C_DIM;
    #pragma unroll
    for (int c = 0; c < C_DIM; ++c) o[c] = row[c] - m - ls;
}

// ---------------------------------------------------------------------------
extern "C" void kernel_launch(void* const* d_in, const int* in_sizes, int n_in,
                              void* d_out, int out_size, void* d_ws, size_t ws_size,
                              hipStream_t stream)
{
    const float* x     = (const float*)d_in[0];
    const int*   edge  = (const int*)d_in[1];
    const float* norm  = (const float*)d_in[2];
    const float* W1    = (const float*)d_in[3];
    const float* b1    = (const float*)d_in[4];
    const float* gamma = (const float*)d_in[5];
    const float* beta  = (const float*)d_in[6];
    const float* mean  = (const float*)d_in[7];
    const float* var   = (const float*)d_in[8];
    const float* W2    = (const float*)d_in[9];
    const float* b2    = (const float*)d_in[10];
    const float* pw    = (const float*)d_in[11];
    const float* pb    = (const float*)d_in[12];

    const int N = in_sizes[0] / F_DIM;
    const int E = in_sizes[2];

    // workspace: h0 | bufA | bufB | acc   (each N*40 fp32, 256B-aligned)
    const size_t per = (((size_t)N * C_DIM) + 63) & ~(size_t)63;
    float* h0   = (float*)d_ws;
    float* bufA = h0   + per;
    float* bufB = bufA + per;
    float* accb = bufB + per;

    const int* src = edge;
    const int* dst = edge + E;

    const long tot     = (long)N * C_DIM;
    const int  zblocks = (int)((tot + 255) / 256);
    const int  nblocks = (N + 255) / 256;

    // 1) fused MLP head (WMMA fp32)
    mlp_kernel<<<(N + 15) / 16, 256, 0, stream>>>(
        x, W1, b1, gamma, beta, mean, var, W2, b2, h0, N);

    // 2) hop-0 contribution
    zero_kernel<<<zblocks, 256, 0, stream>>>(accb, tot);
    combine_kernel<<<nblocks, 256, 0, stream>>>(h0, accb, pw, pb, N);

    // 3) K=10 propagation hops, fused attention-combine (no snapshot tensor)
    const float* prev = h0;
    for (int k = 0; k < 10; ++k) {
        float* next = (k & 1) ? bufB : bufA;
        zero_kernel<<<zblocks, 256, 0, stream>>>(next, tot);
        const long st = (long)E * 8;
        scatter_kernel<<<(int)((st + 255) / 256), 256, 0, stream>>>(
            src, dst, norm, prev, next, E);
        combine_kernel<<<nblocks, 256, 0, stream>>>(next, accb, pw, pb, N);
        prev = next;
    }

    // 4) log_softmax over C=40
    lsm_kernel<<<nblocks, 256, 0, stream>>>(accb, (float*)d_out, N);
}